// ProjSolver_41094247088347
// MI455X (gfx1250) — compile-verified
//
#include <hip/hip_runtime.h>

// Problem dimensions (fixed by the reference)
#define BATCH     2048
#define OUT_DIM   2048   // n
#define M_DIM     1024   // m
#define FREE_NUM_ 1024
#define MAX_ITER_ 16
#define F_TOL_    1e-6f

typedef __attribute__((ext_vector_type(2))) float v2f;
typedef __attribute__((ext_vector_type(4))) float v4f;
typedef __attribute__((ext_vector_type(8))) float v8f;

// GEMM tiling
#define BM 64        // rows of C per workgroup
#define BN 128       // cols of C per workgroup
#define KC 32        // K-chunk staged in LDS
#define LDSS 36      // padded LDS row stride (dwords): 144B rows -> 16B aligned,
                     // 36*m mod 64 distinct for m in [0,16) -> conflict-free b64 reads
#define NCHUNK (OUT_DIM / KC)   // 64 K-chunks

typedef __attribute__((address_space(3))) float lds_f;

// Async DMA: 16 bytes per lane, global memory -> LDS, tracked by ASYNCcnt.
// Bypasses VGPRs entirely (no load-to-register + ds_store round trip).
__device__ __forceinline__ void async_cp16(float* lds_dst, const float* gsrc)
{
    lds_f* ldst = (lds_f*)lds_dst;   // addrspacecast generic->local: 32-bit LDS offset
    asm volatile("global_load_async_to_lds_b128 %0, %1, off"
                 :: "v"(ldst), "v"(gsrc)
                 : "memory");
}

// Stage one K-chunk: Z panel BM x KC (2 x b128/thread) + W panel BN x KC (4 x b128/thread)
__device__ __forceinline__ void stage_chunk(const float* __restrict__ Zin,
                                            const float* __restrict__ Wmat,
                                            int i0, int j0, int k0,
                                            float* zbuf, float* wbuf, int tid)
{
    #pragma unroll
    for (int g = 0; g < 2; ++g) {
        int gi  = tid + g * 256;
        int row = gi >> 3;
        int c4  = (gi & 7) * 4;
        async_cp16(zbuf + row * LDSS + c4,
                   Zin + (size_t)(i0 + row) * OUT_DIM + k0 + c4);
    }
    #pragma unroll
    for (int g = 0; g < 4; ++g) {
        int gi  = tid + g * 256;
        int row = gi >> 3;
        int c4  = (gi & 7) * 4;
        async_cp16(wbuf + row * LDSS + c4,
                   Wmat + (size_t)(j0 + row) * OUT_DIM + k0 + c4);
    }
}

// ---------------------------------------------------------------------------
// Fused WMMA GEMM: C = Zin(BATCHxK) * Wmat(NcolsxK)^T  (both row-major, K=OUT_DIM)
// mode 0: Zout[i][j] = relu-half( acc + vec[j] )        (z-update step)
// mode 1: atomicMax(resmax, max|acc - vec[j]|)          (residual step)
// Double-buffered LDS, async global->LDS staging overlapped with WMMA.
// ---------------------------------------------------------------------------
__global__ __launch_bounds__(256)
void wmma_gemm_kernel(const float* __restrict__ Zin,
                      const float* __restrict__ Wmat,
                      const float* __restrict__ vec,
                      float* __restrict__ Zout,
                      unsigned int* __restrict__ resmax,
                      const int* __restrict__ ctrl,   // ctrl[0] = done flag
                      int mode)
{
    if (ctrl[0]) return;   // uniform early-exit once converged (EXEC stays full)

    __shared__ __align__(16) float Zs[2][BM * LDSS];
    __shared__ __align__(16) float Ws[2][BN * LDSS];

    const int tid  = threadIdx.x;
    const int lane = tid & 31;
    const int wave = tid >> 5;

    const int i0 = blockIdx.y * BM;   // C row base
    const int j0 = blockIdx.x * BN;   // C col base

    const int msub  = wave & 3;          // which 16-row strip (0..3)
    const int nbase = (wave >> 2) * 64;  // 0 or 64: 4 N-subtiles per wave
    const int lm = lane & 15;
    const int kh = lane >> 4;            // K-half for A/B frags, M-half for C

    v8f acc0 = {}, acc1 = {}, acc2 = {}, acc3 = {};

    // Prologue: async-prefetch chunk 0 into buffer 0 (6 in-flight per thread)
    stage_chunk(Zin, Wmat, i0, j0, 0, Zs[0], Ws[0], tid);

    for (int c = 0; c < NCHUNK; ++c) {
        const int cur = c & 1;

        if (c + 1 < NCHUNK) {
            // prefetch next chunk into the other buffer (reads of it finished
            // at the previous iteration's trailing barrier)
            stage_chunk(Zin, Wmat, i0, j0, (c + 1) * KC, Zs[cur ^ 1], Ws[cur ^ 1], tid);
            // async loads retire in order: <=6 outstanding => chunk c landed
            asm volatile("s_wait_asynccnt 0x6" ::: "memory");
        } else {
            asm volatile("s_wait_asynccnt 0x0" ::: "memory");
        }
        __syncthreads();   // all waves' staging for chunk c visible

        const float* Zb = Zs[cur];
        const float* Wb = Ws[cur];

        // --- 8 k-steps of V_WMMA_F32_16X16X4_F32, A reused across 4 N-tiles ---
        #pragma unroll
        for (int kk = 0; kk < KC; kk += 4) {
            // A frag: lane (kh,lm) holds Z[i0+msub*16+lm][k0+kk+2*kh +0/+1]
            v2f a  = *(const v2f*)(Zb + (msub * 16 + lm) * LDSS + kk + 2 * kh);
            // B frags: rows of W are columns of W^T (contiguous K pairs)
            v2f b0 = *(const v2f*)(Wb + (nbase +  0 + lm) * LDSS + kk + 2 * kh);
            v2f b1 = *(const v2f*)(Wb + (nbase + 16 + lm) * LDSS + kk + 2 * kh);
            v2f b2 = *(const v2f*)(Wb + (nbase + 32 + lm) * LDSS + kk + 2 * kh);
            v2f b3 = *(const v2f*)(Wb + (nbase + 48 + lm) * LDSS + kk + 2 * kh);
            acc0 = __builtin_amdgcn_wmma_f32_16x16x4_f32(false, a, false, b0, (short)0, acc0, false, false);
            acc1 = __builtin_amdgcn_wmma_f32_16x16x4_f32(false, a, false, b1, (short)0, acc1, false, false);
            acc2 = __builtin_amdgcn_wmma_f32_16x16x4_f32(false, a, false, b2, (short)0, acc2, false, false);
            acc3 = __builtin_amdgcn_wmma_f32_16x16x4_f32(false, a, false, b3, (short)0, acc3, false, false);
        }
        __syncthreads();   // reads of buf[cur] done -> safe to prefetch into it next iter
    }

    // --- epilogue ---
    const int mrow = i0 + msub * 16;
    if (mode == 0) {
        #pragma unroll
        for (int s = 0; s < 4; ++s) {
            v8f acc = (s == 0) ? acc0 : (s == 1) ? acc1 : (s == 2) ? acc2 : acc3;
            int j = j0 + nbase + s * 16 + lm;   // C/D: N = lane&15
            float bj = vec[j];
            #pragma unroll
            for (int r = 0; r < 8; ++r) {
                int m = mrow + r + 8 * kh;      // C/D: M = reg + 8*(lane>>4)
                float v = acc[r] + bj;
                if (j >= FREE_NUM_) v = fmaxf(v, 0.0f);
                Zout[(size_t)m * OUT_DIM + j] = v;
            }
        }
    } else {
        float mymax = 0.0f;
        #pragma unroll
        for (int s = 0; s < 4; ++s) {
            v8f acc = (s == 0) ? acc0 : (s == 1) ? acc1 : (s == 2) ? acc2 : acc3;
            int j = j0 + nbase + s * 16 + lm;
            float bj = vec[j];
            #pragma unroll
            for (int r = 0; r < 8; ++r)
                mymax = fmaxf(mymax, fabsf(acc[r] - bj));
        }
        // wave32 max-reduction, one atomic per wave (float bits monotone for >=0)
        #pragma unroll
        for (int off = 16; off > 0; off >>= 1)
            mymax = fmaxf(mymax, __shfl_xor(mymax, off, 32));
        if (lane == 0) atomicMax(resmax, __float_as_uint(mymax));
    }
}

// bias[j] = sum_k b[k] * WbProj[j][k]   (one wave per output row)
__global__ __launch_bounds__(256)
void bias_kernel(const float* __restrict__ b, const float* __restrict__ Wb,
                 float* __restrict__ bias)
{
    int wave = threadIdx.x >> 5, lane = threadIdx.x & 31;
    int j = blockIdx.x * 8 + wave;
    const float* row = Wb + (size_t)j * M_DIM;
    float s = 0.0f;
    for (int k = lane; k < M_DIM; k += 32) s += row[k] * b[k];
    #pragma unroll
    for (int off = 16; off > 0; off >>= 1) s += __shfl_xor(s, off, 32);
    if (lane == 0) bias[j] = s;
}

// copy input z -> z_cur (d_out), zero control words
__global__ __launch_bounds__(256)
void init_copy_kernel(const float* __restrict__ z, float* __restrict__ zcur,
                      int* __restrict__ ctrl, unsigned int* __restrict__ res)
{
    size_t idx = (size_t)blockIdx.x * blockDim.x + threadIdx.x;
    ((v4f*)zcur)[idx] = ((const v4f*)z)[idx];
    if (idx == 0) { ctrl[0] = 0; ctrl[1] = 0; *res = 0u; }
}

// z_cur = done ? z_cur : z_new   (reads the *old* done flag)
__global__ __launch_bounds__(256)
void update_z_kernel(const float* __restrict__ znew, float* __restrict__ zcur,
                     const int* __restrict__ ctrl)
{
    if (ctrl[0]) return;
    size_t idx = (size_t)blockIdx.x * blockDim.x + threadIdx.x;
    ((v4f*)zcur)[idx] = ((const v4f*)znew)[idx];
}

// iters += !done; done |= (res <= tol); res = 0; publish curr_iter = iters+1
__global__ void finalize_kernel(int* __restrict__ ctrl, unsigned int* __restrict__ res,
                                float* __restrict__ out_iter)
{
    int done = ctrl[0];
    if (!done) ctrl[1] += 1;
    float r = __uint_as_float(*res);
    if (r <= F_TOL_) ctrl[0] = 1;
    *res = 0u;
    *out_iter = (float)(ctrl[1] + 1);
}

extern "C" void kernel_launch(void* const* d_in, const int* in_sizes, int n_in,
                              void* d_out, int out_size, void* d_ws, size_t ws_size,
                              hipStream_t stream)
{
    (void)in_sizes; (void)n_in; (void)out_size; (void)ws_size;
    const float* z  = (const float*)d_in[0];
    const float* b  = (const float*)d_in[1];
    const float* A  = (const float*)d_in[2];
    const float* Wz = (const float*)d_in[3];
    const float* Wb = (const float*)d_in[4];

    float* out      = (float*)d_out;
    float* zcur     = out;                                  // BATCH*OUT_DIM floats
    float* out_iter = out + (size_t)BATCH * OUT_DIM;        // last output element

    char* ws = (char*)d_ws;
    float* znew = (float*)ws;                                                // 16 MB
    float* biasv = (float*)(ws + (size_t)BATCH * OUT_DIM * sizeof(float));   // 8 KB
    int* ctrl = (int*)(biasv + OUT_DIM);                                     // done, iters
    unsigned int* res = (unsigned int*)(ctrl + 2);

    const int nElemVec4 = (BATCH * OUT_DIM) / 4;            // 1M float4
    init_copy_kernel<<<nElemVec4 / 256, 256, 0, stream>>>(z, zcur, ctrl, res);
    bias_kernel<<<OUT_DIM / 8, 256, 0, stream>>>(b, Wb, biasv);

    dim3 blk(256);
    dim3 gZ(OUT_DIM / BN, BATCH / BM);   // (16, 32): z-update GEMM
    dim3 gR(M_DIM  / BN, BATCH / BM);    // ( 8, 32): residual GEMM

    for (int it = 0; it < MAX_ITER_; ++it) {
        wmma_gemm_kernel<<<gZ, blk, 0, stream>>>(zcur, Wz, biasv, znew, nullptr, ctrl, 0);
        wmma_gemm_kernel<<<gR, blk, 0, stream>>>(znew, A, b, nullptr, res, ctrl, 1);
        update_z_kernel<<<nElemVec4 / 256, 256, 0, stream>>>(znew, zcur, ctrl);
        finalize_kernel<<<1, 1, 0, stream>>>(ctrl, res, out_iter);
    }
}